// TaylorExp_27891517621028
// MI455X (gfx1250) — compile-verified
//
#include <hip/hip_runtime.h>
#include <hip/hip_bf16.h>

typedef float v2f __attribute__((ext_vector_type(2)));
typedef float v8f __attribute__((ext_vector_type(8)));

#define INPUT_DIM 16
#define OUT_DIM 273   // 1 + 16 + 256

// Per row r of x (16 floats):
//   out[r][0]        = 1
//   out[r][1..16]    = x * 0.5                        (1/16^(1/4))
//   out[r][17..272]  = (x x^T) * 1/(sqrt(2)*sqrt(16)) row-major
//
// The 16x16 outer product is computed as a rank-1 V_WMMA_F32_16X16X4_F32:
// A(16x4) has x in column K=0 (pre-scaled), B(4x16) has x in row K=0.
__global__ __launch_bounds__(256) void TaylorExp_kernel(const float* __restrict__ x,
                                                        float* __restrict__ out,
                                                        int n_rows) {
    const int lane = threadIdx.x & 31;
    const int wave_in_block = threadIdx.x >> 5;
    const int waves_per_block = blockDim.x >> 5;
    const int wave_id = blockIdx.x * waves_per_block + wave_in_block;
    const int n_waves = gridDim.x * waves_per_block;

    const float SQ = 0.17677669529663687f;   // 1/(sqrt(2)*sqrt(16))
    const float SL = 0.5f;                   // 1/16^(1/4)

    const bool lo = lane < 16;
    const int m = lane & 15;
    // Quadratic-block per-lane dword offset for C VGPR i:
    //   lanes 0-15  -> row M=i,   col=lane      -> i*16 + lane
    //   lanes 16-31 -> row M=i+8, col=lane-16   -> i*16 + 128 + lane - 16 = i*16 + lane + 112
    const int qoff = lane + (lo ? 0 : 112);

    for (int row = wave_id; row < n_rows; row += n_waves) {
        const float* xr = x + (size_t)row * INPUT_DIM;
        const float v = xr[m];               // lanes 16-31 re-read lanes 0-15's line (same 64B)

        // Rank-1 WMMA operands: only K=0 column/row populated, upper half-wave zero.
        v2f a; a.x = lo ? v * SQ : 0.0f; a.y = 0.0f;   // A 16x4: lane<16 VGPR0 = K=0
        v2f b; b.x = lo ? v      : 0.0f; b.y = 0.0f;   // B 4x16: lane<16 VGPR0 = K=0
        v8f c = {};
        c = __builtin_amdgcn_wmma_f32_16x16x4_f32(
                /*neg_a=*/false, a, /*neg_b=*/false, b,
                /*c_mod=*/(short)0, c, /*reuse_a=*/false, /*reuse_b=*/false);

        float* o = out + (size_t)row * OUT_DIM;

        // t0 and t1 (small, predicated lanes)
        if (lane == 0) {
            __builtin_nontemporal_store(1.0f, o);
        }
        if (lo) {
            __builtin_nontemporal_store(v * SL, o + 1 + lane);
        }

        // quadratic block: 8 coalesced b32 stores (each covers 2 contiguous 64B runs)
        float* oq = o + 17 + qoff;
#pragma unroll
        for (int i = 0; i < 8; ++i) {
            __builtin_nontemporal_store(c[i], oq + (size_t)(i * 16));
        }
    }
}

extern "C" void kernel_launch(void* const* d_in, const int* in_sizes, int n_in,
                              void* d_out, int out_size, void* d_ws, size_t ws_size,
                              hipStream_t stream) {
    const float* x = (const float*)d_in[0];
    float* out = (float*)d_out;

    const int n_rows = in_sizes[0] / INPUT_DIM;   // 262144 for the reference shapes

    // 8 waves per 256-thread block; target ~8 rows per wave to amortize setup.
    const int waves_per_block = 256 / 32;
    int target_waves = (n_rows + 7) / 8;
    int blocks = (target_waves + waves_per_block - 1) / waves_per_block;
    if (blocks < 1) blocks = 1;
    if (blocks > 65535) blocks = 65535;

    TaylorExp_kernel<<<blocks, 256, 0, stream>>>(x, out, n_rows);
}